// SGSELayer_11312943858098
// MI455X (gfx1250) — compile-verified
//
#include <hip/hip_runtime.h>
#include <math.h>

// MI455X / gfx1250. Bandwidth-bound squeeze-excite:
//   pass1: stream x once, compute all 12 group sums per (b,c)      (256 MB read)
//   pass2: tiny 64->4->64 MLP per (dim,b,g) via V_WMMA_F32_16X16X4_F32
//   pass3: stream x, scale by (sd+sh+sw)/3, write out              (512 MB r+w)
// ~768 MB @ 23.3 TB/s => ~33 us floor.

typedef float v2f __attribute__((ext_vector_type(2)));
typedef float v8f __attribute__((ext_vector_type(8)));

#define CB 4      // batch
#define CCH 64    // channels
#define CL 64     // D=H=W
#define CG 4      // groups
#define CHID 4    // hidden

// ---------------------------------------------------------------------------
// Kernel 1: fused pooling. One block per (b,c); 256 threads stream the
// 64^3 slice as float4 and accumulate 12 group sums (4 per spatial dim).
// Index trick: float4-flat index f = i*256 + tid  (i = iteration)
//   w4 = tid[3:0]  -> w-group = tid[3:2]           (constant per thread)
//   h  = {i[1:0], tid[7:4]} -> h-group = i[1:0]    (static in unrolled loop)
//   d  = i[7:2]            -> d-group = i[7:6]     (static in unrolled loop)
// so all group indices are compile-time static => no dynamic reg indexing.
// ---------------------------------------------------------------------------
__global__ void sgse_pool_kernel(const float* __restrict__ x,
                                 float* __restrict__ sums /* [B*C][12] */) {
    const int bc  = blockIdx.x;           // b*64 + c
    const int tid = threadIdx.x;          // 0..255
    const float4* xs = (const float4*)(x + (size_t)bc * (CL * CL * CL));

    float hsum[4] = {0.f, 0.f, 0.f, 0.f};
    float dsum[4] = {0.f, 0.f, 0.f, 0.f};

#pragma unroll
    for (int dg = 0; dg < 4; ++dg) {
        float ds = 0.f;
#pragma unroll
        for (int hg = 0; hg < 4; ++hg) {
            float hs = 0.f;
#pragma unroll 4
            for (int j = 0; j < 16; ++j) {
                const int f = (dg * 64 + j * 4 + hg) * 256 + tid;
                const float4 v = xs[f];
                hs += (v.x + v.y) + (v.z + v.w);
            }
            hsum[hg] += hs;
            ds += hs;
        }
        dsum[dg] += ds;
    }
    const float wtot = (dsum[0] + dsum[1]) + (dsum[2] + dsum[3]);
    const int wg = (tid >> 2) & 3;        // w-group for this thread's lane

    __shared__ float acc12[12];
    if (tid < 12) acc12[tid] = 0.f;
    __syncthreads();
#pragma unroll
    for (int g = 0; g < 4; ++g) {
        atomicAdd(&acc12[0 * 4 + g], dsum[g]);   // ds_add_f32
        atomicAdd(&acc12[1 * 4 + g], hsum[g]);
    }
    atomicAdd(&acc12[2 * 4 + wg], wtot);
    __syncthreads();
    if (tid < 12) sums[bc * 12 + tid] = acc12[tid];
}

// ---------------------------------------------------------------------------
// Kernel 2: the squeeze-excite MLP, one wave per spatial dim (blockIdx.x).
// GEMM1: Y[16x64] @ W1[64x4]  -> 16 x V_WMMA_F32_16X16X4_F32 (K-steps of 4)
// GEMM2: H[16x4]  @ W2[4x64]  -> 4  x V_WMMA_F32_16X16X4_F32 (N-tiles of 16)
// H is re-staged D-layout -> A-layout through 256 B of LDS.
// Rows M = b*4+g (16 of them), EXEC all-1s (blockDim=32, no divergence at WMMA).
// ---------------------------------------------------------------------------
__global__ void sgse_mlp_kernel(const float* __restrict__ sums,
                                const float* __restrict__ w1,   // [2][3][64][4]
                                const float* __restrict__ b1,   // [2][3][4]
                                const float* __restrict__ w2,   // [2][3][4][64]
                                const float* __restrict__ b2,   // [2][3][64]
                                const int*   __restrict__ cls_ptr,
                                float* __restrict__ scales /* [3][B][C][4] */) {
    const int dim  = blockIdx.x;          // 0..2
    const int lane = threadIdx.x;         // 0..31
    const int lh   = lane >> 4;           // lane half
    const int mr   = lane & 15;           // A-row index / B-col index
    const int cls  = cls_ptr[0];
    const float inv = 1.0f / 65536.0f;    // 16*64*64 elements per pooled slab

    const float* W1 = w1 + (size_t)(cls * 3 + dim) * CCH * CHID;
    const float* B1 = b1 + (cls * 3 + dim) * CHID;
    const float* W2 = w2 + (size_t)(cls * 3 + dim) * CHID * CCH;
    const float* B2 = b2 + (cls * 3 + dim) * CCH;

    const int bi = mr >> 2;               // batch of this row
    const int gi = mr & 3;                // group of this row

    // ---- GEMM1: accumulate over K = C = 64 in steps of 4 -------------------
    v8f acc = {0.f, 0.f, 0.f, 0.f, 0.f, 0.f, 0.f, 0.f};
#pragma unroll
    for (int kk = 0; kk < 16; ++kk) {
        const int k0 = kk * 4 + lh * 2;   // this lane's K pair
        v2f a, bmat;
        a.x = sums[(bi * 64 + (k0 + 0)) * 12 + dim * 4 + gi] * inv;  // Y[mr,k0]
        a.y = sums[(bi * 64 + (k0 + 1)) * 12 + dim * 4 + gi] * inv;  // Y[mr,k0+1]
        bmat.x = (mr < CHID) ? W1[(k0 + 0) * CHID + mr] : 0.0f;      // W1[k0,  mr]
        bmat.y = (mr < CHID) ? W1[(k0 + 1) * CHID + mr] : 0.0f;      // W1[k0+1,mr]
        acc = __builtin_amdgcn_wmma_f32_16x16x4_f32(
            /*neg_a=*/false, a, /*neg_b=*/false, bmat,
            /*c_mod=*/(short)0, acc, /*reuse_a=*/false, /*reuse_b=*/false);
    }

    // ---- bias + relu, stage hidden matrix H[16][4] via LDS -----------------
    __shared__ float hlds[16 * CHID];
    if (mr < CHID) {
        const float bias = B1[mr];
#pragma unroll
        for (int j = 0; j < 8; ++j) {
            const int row = lh ? (j + 8) : j;     // D layout: VGPR j, halves
            float v = acc[j] + bias;
            hlds[row * CHID + mr] = v > 0.f ? v : 0.f;
        }
    }
    __syncthreads();

    // ---- GEMM2: A = H in A-layout, loop 4 N-tiles of 16 channels ----------
    const int k0 = lh * 2;
    v2f a2;
    a2.x = hlds[mr * CHID + k0];
    a2.y = hlds[mr * CHID + k0 + 1];

#pragma unroll
    for (int t = 0; t < 4; ++t) {
        const int col = t * 16 + mr;              // output channel
        v2f b2v;
        b2v.x = W2[(k0 + 0) * CCH + col];
        b2v.y = W2[(k0 + 1) * CCH + col];
        v8f s = {0.f, 0.f, 0.f, 0.f, 0.f, 0.f, 0.f, 0.f};
        s = __builtin_amdgcn_wmma_f32_16x16x4_f32(
            false, a2, false, b2v, (short)0, s, false, false);
        const float bias2 = B2[col];
#pragma unroll
        for (int j = 0; j < 8; ++j) {
            const int row = lh ? (j + 8) : j;     // row = b*4+g
            const float v = s[j] + bias2;
            const float sig = 1.0f / (1.0f + expf(-v));
            scales[((dim * CB + (row >> 2)) * CCH + col) * CG + (row & 3)] = sig;
        }
    }
}

// ---------------------------------------------------------------------------
// Kernel 3: apply gates. One float4 per thread, perfectly coalesced.
// float4-flat index bits: [3:0]=w4, [9:4]=h, [15:10]=d, [21:16]=c, [23:22]=b.
// Within a float4 (4 consecutive w) the w-group is constant (4 | 16).
// Scale table is 12 KB -> L2/WGP$ resident, effectively free to read.
// ---------------------------------------------------------------------------
__global__ void sgse_apply_kernel(const float* __restrict__ x,
                                  const float* __restrict__ scales,
                                  float* __restrict__ out) {
    const int f  = blockIdx.x * 256 + threadIdx.x;  // float4 index
    const int w4 = f & 15;
    const int h  = (f >> 4) & 63;
    const int d  = (f >> 10) & 63;
    const int c  = (f >> 16) & 63;
    const int b  = (f >> 22);

    const float sd = scales[((0 * CB + b) * CCH + c) * CG + (d >> 4)];
    const float sh = scales[((1 * CB + b) * CCH + c) * CG + (h >> 4)];
    const float sw = scales[((2 * CB + b) * CCH + c) * CG + (w4 >> 2)];
    const float m = (sd + sh + sw) * (1.0f / 3.0f);

    const float4 v = ((const float4*)x)[f];
    float4 o;
    o.x = v.x * m; o.y = v.y * m; o.z = v.z * m; o.w = v.w * m;
    ((float4*)out)[f] = o;
}

// ---------------------------------------------------------------------------
extern "C" void kernel_launch(void* const* d_in, const int* in_sizes, int n_in,
                              void* d_out, int out_size, void* d_ws, size_t ws_size,
                              hipStream_t stream) {
    const float* x   = (const float*)d_in[0];
    const float* w1  = (const float*)d_in[1];
    const float* b1  = (const float*)d_in[2];
    const float* w2  = (const float*)d_in[3];
    const float* b2  = (const float*)d_in[4];
    const int*   cls = (const int*)d_in[5];
    float* out = (float*)d_out;

    float* sums   = (float*)d_ws;           // 256*12 floats = 12 KB
    float* scales = sums + CB * CCH * 12;   // 3*4*64*4 floats = 12 KB

    // Pass 1: pooling (one block per (b,c))
    sgse_pool_kernel<<<CB * CCH, 256, 0, stream>>>(x, sums);
    // Pass 2: WMMA MLP (one wave per spatial dim)
    sgse_mlp_kernel<<<3, 32, 0, stream>>>(sums, w1, b1, w2, b2, cls, scales);
    // Pass 3: gate + average (16M float4s)
    const int total4 = CB * CCH * CL * CL * (CL / 4);   // 16,777,216
    sgse_apply_kernel<<<total4 / 256, 256, 0, stream>>>(x, scales, out);
}